// GraphConvolutionalNetwork_20633022890139
// MI455X (gfx1250) — compile-verified
//
#include <hip/hip_runtime.h>

// Problem constants (from reference)
#define NN      100000
#define EE      1600000
#define F_IN    7
#define H_LOW   100
#define H_HIGH  200
#define F_OUT   7
#define N_PRED  10000

typedef __attribute__((ext_vector_type(2))) float v2f;
typedef __attribute__((ext_vector_type(8))) float v8f;

// ---------------------------------------------------------------------------
// Degree / norm kernels
// ---------------------------------------------------------------------------
__global__ void init_deg_kernel(float* __restrict__ deg) {
    int i = blockIdx.x * blockDim.x + threadIdx.x;
    if (i < NN) deg[i] = 1.0f;  // self-loop contributes 1 to every node
}

__global__ void deg_accum_kernel(float* __restrict__ deg, const int* __restrict__ dst) {
    int e = blockIdx.x * blockDim.x + threadIdx.x;
    if (e < EE) atomicAdd(&deg[dst[e]], 1.0f);
}

__global__ void rsqrt_kernel(float* __restrict__ deg) {
    int i = blockIdx.x * blockDim.x + threadIdx.x;
    if (i < NN) deg[i] = rsqrtf(deg[i]);   // in place: deg -> dinv
}

__global__ void zero4_kernel(float4* __restrict__ p, long long n4) {
    long long t = (long long)blockIdx.x * blockDim.x + threadIdx.x;
    if (t < n4) p[t] = make_float4(0.f, 0.f, 0.f, 0.f);
}

// ---------------------------------------------------------------------------
// WMMA GEMM: out[M x Nc] = op(A[M x K] @ W[K x Nc] (+ bias) (relu))
// One wave32 per 16x16 output tile, V_WMMA_F32_16X16X4_F32, K-step 4.
// K is compile-time -> fully unrolled MAC loop, no runtime K compares.
// All loads are unconditional with clamped addresses (no EXEC branching):
//   - A tail elements (K=7 case) are zeroed with a post-load select.
//   - B columns beyond Nc are loaded clamped; they only feed D columns that
//     are never stored, and 0 * finite == 0 keeps valid columns exact.
// M must be a multiple of 16.
// ---------------------------------------------------------------------------
template <int K, bool RELU, bool BIAS>
__global__ void wmma_gemm_f32_kernel(const float* __restrict__ A,
                                     const float* __restrict__ W,
                                     const float* __restrict__ bias,
                                     float* __restrict__ out,
                                     int M, int Nc, int nColTiles) {
    const int warp = threadIdx.x >> 5;
    const int lane = threadIdx.x & 31;
    const long long tile   = (long long)blockIdx.x * (blockDim.x >> 5) + warp;
    const long long nTiles = (long long)(M >> 4) * nColTiles;
    if (tile >= nTiles) return;              // wave-uniform: EXEC stays all-ones

    const int rowTile = (int)(tile / nColTiles);
    const int colTile = (int)(tile % nColTiles);
    const int rowBase = rowTile << 4;
    const int colBase = colTile << 4;
    const int halfsel = lane >> 4;           // 0: K offsets {0,1}; 1: {2,3}
    const int lane16  = lane & 15;
    const int kBase   = halfsel << 1;

    const float* __restrict__ Arow = A + (long long)(rowBase + lane16) * K;

    const int  col  = colBase + lane16;      // B/D column this lane feeds
    const bool colOK = (col < Nc);
    const int  colc  = colOK ? col : (Nc - 1);          // clamped (always valid)
    const float* __restrict__ Wcol = W + colc;

    v8f c = {0.f, 0.f, 0.f, 0.f, 0.f, 0.f, 0.f, 0.f};
#pragma unroll
    for (int k = 0; k < K; k += 4) {
        const int k0  = k + kBase;
        const int k1  = k0 + 1;
        const int k0c = (k0 < K) ? k0 : (K - 1);        // clamped addresses:
        const int k1c = (k1 < K) ? k1 : (K - 1);        // loads always in-bounds
        const float a0 = Arow[k0c];
        const float a1 = Arow[k1c];
        v2f a, b;
        a.x = (k0 < K) ? a0 : 0.f;   // constant-folds except in the K tail
        a.y = (k1 < K) ? a1 : 0.f;
        b.x = Wcol[(long long)k0c * Nc];
        b.y = Wcol[(long long)k1c * Nc];
        // 8 args: (neg_a, A, neg_b, B, c_mod, C, reuse_a, reuse_b)
        c = __builtin_amdgcn_wmma_f32_16x16x4_f32(false, a, false, b,
                                                  (short)0, c, false, false);
    }

    if (colOK) {
        float bv = 0.f;
        if (BIAS) bv = bias[col];
        const int rbase = rowBase + (halfsel << 3);  // rows rbase..rbase+7
#pragma unroll
        for (int r = 0; r < 8; ++r) {
            float v = c[r] + bv;
            if (RELU) v = fmaxf(v, 0.f);
            out[(long long)(rbase + r) * Nc + col] = v;
        }
    }
}

// ---------------------------------------------------------------------------
// Edge scatter: agg[dst] += h[src] * dinv[src]*dinv[dst]  (H_LOW = 100 feats)
// One wave per edge; lanes 0..24 each move a float4 (25*4 = 100 features).
// h/agg (40 MB each) are L2-resident on MI455X (192 MB L2); atomics resolve
// in L2 (non-returning global_atomic_add_f32).
// ---------------------------------------------------------------------------
__global__ void scatter_edges_kernel(const float* __restrict__ h,
                                     float* __restrict__ agg,
                                     const int* __restrict__ src,
                                     const int* __restrict__ dst,
                                     const float* __restrict__ dinv) {
    const int warp = threadIdx.x >> 5;
    const int lane = threadIdx.x & 31;
    const long long e = (long long)blockIdx.x * (blockDim.x >> 5) + warp;
    if (e >= EE) return;

    const int s = src[e];
    const int d = dst[e];
    const float norm = dinv[s] * dinv[d];

    if (lane < (H_LOW / 4)) {
        const float4* __restrict__ hrow =
            (const float4*)(h + (long long)s * H_LOW);
        float4 v = hrow[lane];
        float* __restrict__ o = agg + (long long)d * H_LOW + (lane << 2);
        atomicAdd(o + 0, v.x * norm);
        atomicAdd(o + 1, v.y * norm);
        atomicAdd(o + 2, v.z * norm);
        atomicAdd(o + 3, v.w * norm);
    }
}

// ---------------------------------------------------------------------------
// Layer epilogue: agg = relu(agg + h_dense * dinv^2 (self-loop) + bias)
// ---------------------------------------------------------------------------
__global__ void epilogue_kernel(float* __restrict__ agg,
                                const float* __restrict__ hdense,
                                const float* __restrict__ dinv,
                                const float* __restrict__ bias) {
    const long long t = (long long)blockIdx.x * blockDim.x + threadIdx.x;
    if (t >= (long long)NN * H_LOW) return;
    const int i = (int)(t / H_LOW);
    const int f = (int)(t % H_LOW);
    const float di = dinv[i];
    const float v = agg[t] + hdense[t] * di * di + bias[f];
    agg[t] = fmaxf(v, 0.f);
}

// ---------------------------------------------------------------------------
// Host launcher
// ---------------------------------------------------------------------------
extern "C" void kernel_launch(void* const* d_in, const int* in_sizes, int n_in,
                              void* d_out, int out_size, void* d_ws, size_t ws_size,
                              hipStream_t stream) {
    const float* x  = (const float*)d_in[0];
    const int*   ei = (const int*)d_in[1];   // (2, E): row0=src, row1=dst
    const float* W1 = (const float*)d_in[2];
    const float* b1 = (const float*)d_in[3];
    const float* W2 = (const float*)d_in[4];
    const float* b2 = (const float*)d_in[5];
    const float* W3 = (const float*)d_in[6];
    const float* b3 = (const float*)d_in[7];
    const float* W4 = (const float*)d_in[8];
    const float* b4 = (const float*)d_in[9];
    float* out = (float*)d_out;

    const int* src = ei;
    const int* dst = ei + EE;

    // Workspace layout (bytes): dinv[N] | bufA[N*100] | bufB[N*100]
    char* ws = (char*)d_ws;
    float* dinv = (float*)ws;
    float* bufA = (float*)(ws + (size_t)NN * 4);
    float* bufB = (float*)(ws + (size_t)NN * 4 + (size_t)NN * H_LOW * 4);
    float* bufC = bufA;  // 10000x200 fits in bufA after layer-2 scatter consumes it

    const int B256 = 256;

    // --- degree / normalization ---
    init_deg_kernel<<<(NN + B256 - 1) / B256, B256, 0, stream>>>(dinv);
    deg_accum_kernel<<<(EE + B256 - 1) / B256, B256, 0, stream>>>(dinv, dst);
    rsqrt_kernel<<<(NN + B256 - 1) / B256, B256, 0, stream>>>(dinv);

    const long long n4   = (long long)NN * H_LOW / 4;
    const int zeroBlocks = (int)((n4 + B256 - 1) / B256);
    const int scatBlocks = (EE + 7) / 8;                       // 8 waves / block
    const long long nElem = (long long)NN * H_LOW;
    const int epiBlocks  = (int)((nElem + B256 - 1) / B256);

    auto gemmBlocks = [](int M, int colTiles) {
        long long nTiles = (long long)(M >> 4) * colTiles;
        return (int)((nTiles + 7) / 8);                        // 8 waves / block
    };

    // --- layer 1: hA = x @ W1 ; agg ; h1 = relu(agg + self + b1) ---
    wmma_gemm_f32_kernel<F_IN, false, false><<<gemmBlocks(NN, 7), B256, 0, stream>>>(
        x, W1, nullptr, bufA, NN, H_LOW, 7);
    zero4_kernel<<<zeroBlocks, B256, 0, stream>>>((float4*)bufB, n4);
    scatter_edges_kernel<<<scatBlocks, B256, 0, stream>>>(bufA, bufB, src, dst, dinv);
    epilogue_kernel<<<epiBlocks, B256, 0, stream>>>(bufB, bufA, dinv, b1);

    // --- layer 2: hA = h1 @ W2 ; agg ; h2 = relu(agg + self + b2) ---
    wmma_gemm_f32_kernel<H_LOW, false, false><<<gemmBlocks(NN, 7), B256, 0, stream>>>(
        bufB, W2, nullptr, bufA, NN, H_LOW, 7);
    zero4_kernel<<<zeroBlocks, B256, 0, stream>>>((float4*)bufB, n4);
    scatter_edges_kernel<<<scatBlocks, B256, 0, stream>>>(bufA, bufB, src, dst, dinv);
    epilogue_kernel<<<epiBlocks, B256, 0, stream>>>(bufB, bufA, dinv, b2);

    // --- layer 3: hC = relu(h2[last 10k] @ W3 + b3) ---
    const float* h2tail = bufB + (long long)(NN - N_PRED) * H_LOW;
    wmma_gemm_f32_kernel<H_LOW, true, true><<<gemmBlocks(N_PRED, 13), B256, 0, stream>>>(
        h2tail, W3, b3, bufC, N_PRED, H_HIGH, 13);

    // --- layer 4: out = hC @ W4 + b4 ---
    wmma_gemm_f32_kernel<H_HIGH, false, true><<<gemmBlocks(N_PRED, 1), B256, 0, stream>>>(
        bufC, W4, b4, out, N_PRED, F_OUT, 1);
}